// MoELayer_12773232738933
// MI455X (gfx1250) — compile-verified
//
#include <hip/hip_runtime.h>

#define D_MODEL 512
#define D_FF    512
#define N_EXP   16
#define TOPK    3
#define T_TOK   2048
#define NSLOT   (T_TOK * TOPK)
#define MTILE   32
#define BLOCK   512          // 16 waves
#define LDSPAD  516          // 512 + 4: 4*r mod 64 distinct for r<16 -> conflict-free frags

typedef float v2f __attribute__((ext_vector_type(2)));
typedef float v4f __attribute__((ext_vector_type(4)));
typedef float v8f __attribute__((ext_vector_type(8)));

__device__ __forceinline__ v8f wmma4(v2f a, v2f b, v8f c) {
  // D = A(16x4) * B(4x16) + C, fp32 tensor op (VOP3P V_WMMA_F32_16X16X4_F32)
  return __builtin_amdgcn_wmma_f32_16x16x4_f32(false, a, false, b, (short)0, c,
                                               false, false);
}

struct TileSmem {
  float sX[MTILE][LDSPAD];
  float sH[MTILE][LDSPAD];
  int   sTok[MTILE];
  int   sDst[MTILE];
};

// Fused 32-token SwiGLU FFN tile: dst[sDst[m]] = swiglu(X[sTok[m]] @ Wgu^T) @ Wd^T
// 16 waves; each wave owns a 16-wide N subtile and both 16-row M fragments,
// so every B fragment load feeds 2 WMMAs (4 in pass A with gate+up fused).
__device__ __forceinline__ void tile_ffn(TileSmem& sm,
                                         const float* __restrict__ x,
                                         const float* __restrict__ wgu,  // [2F, D]
                                         const float* __restrict__ wd,   // [D, F]
                                         float* __restrict__ dst)        // rows of D_MODEL
{
  const int tid   = threadIdx.x;
  const int lane  = tid & 31;
  const int wave  = tid >> 5;
  const int hl    = lane & 15;          // half-lane: row (A) / col (B)
  const int koff  = (lane >> 4) << 1;   // 0 or 2 : K sub-offset per ISA layout
  const int rbase = (lane < 16) ? 0 : 8;

  // ---- stage gathered 32x512 X tile into LDS (float4, fully coalesced) ----
  {
    const int cg = (tid & 127) << 2;    // 0..508
    const int rg = tid >> 7;            // 0..3
    #pragma unroll
    for (int i = 0; i < 8; ++i) {
      const int r = i * 4 + rg;
      const v4f v = *(const v4f*)(x + (size_t)sm.sTok[r] * D_MODEL + cg);
      *(v4f*)&sm.sX[r][cg] = v;
    }
  }
  __syncthreads();

  // ---- pass A: gate & up together, two M fragments, SwiGLU in registers ----
  for (int it = 0; it < D_FF / 256; ++it) {
    const int f = it * 256 + wave * 16 + hl;              // 0..511
    const float* wg = wgu + (size_t)f * D_MODEL;          // gate row
    const float* wu = wgu + (size_t)(D_FF + f) * D_MODEL; // up row
    v8f accG0 = {}, accU0 = {}, accG1 = {}, accU1 = {};
    for (int kb = 0; kb < D_MODEL; kb += 4) {
      v2f a0 = *(const v2f*)&sm.sX[hl][kb + koff];
      v2f a1 = *(const v2f*)&sm.sX[16 + hl][kb + koff];
      v2f bg = *(const v2f*)(wg + kb + koff);
      v2f bu = *(const v2f*)(wu + kb + koff);
      accG0 = wmma4(a0, bg, accG0);
      accU0 = wmma4(a0, bu, accU0);
      accG1 = wmma4(a1, bg, accG1);
      accU1 = wmma4(a1, bu, accU1);
    }
    #pragma unroll
    for (int r = 0; r < 8; ++r) {
      float g0 = accG0[r];
      float g1 = accG1[r];
      sm.sH[rbase + r][f]      = (g0 / (1.0f + __expf(-g0))) * accU0[r];
      sm.sH[16 + rbase + r][f] = (g1 / (1.0f + __expf(-g1))) * accU1[r];
    }
  }
  __syncthreads();

  // ---- pass B: O = H @ Wd^T ----
  for (int it = 0; it < D_MODEL / 256; ++it) {
    const int dcol = it * 256 + wave * 16 + hl;
    const float* wr = wd + (size_t)dcol * D_FF;
    v8f acc0 = {}, acc1 = {};
    for (int kb = 0; kb < D_FF; kb += 4) {
      v2f a0 = *(const v2f*)&sm.sH[hl][kb + koff];
      v2f a1 = *(const v2f*)&sm.sH[16 + hl][kb + koff];
      v2f b  = *(const v2f*)(wr + kb + koff);
      acc0 = wmma4(a0, b, acc0);
      acc1 = wmma4(a1, b, acc1);
    }
    #pragma unroll
    for (int r = 0; r < 8; ++r) {
      int d0 = sm.sDst[rbase + r];
      int d1 = sm.sDst[16 + rbase + r];
      if (d0 >= 0) dst[(size_t)d0 * D_MODEL + dcol] = acc0[r];
      if (d1 >= 0) dst[(size_t)d1 * D_MODEL + dcol] = acc1[r];
    }
  }
  __syncthreads();
}

// ---------------- router: sigmoid affinity, bias-adjusted top-3 ----------------
__global__ __launch_bounds__(256) void router_kernel(
    const float* __restrict__ x, const float* __restrict__ gw,
    const float* __restrict__ bias,
    int* __restrict__ sel, float* __restrict__ gating, int* __restrict__ counts)
{
  const int lane = threadIdx.x & 31;
  const int wave = threadIdx.x >> 5;
  const int t = blockIdx.x * 8 + wave;
  if (t >= T_TOK) return;

  float acc[N_EXP];
  #pragma unroll
  for (int e = 0; e < N_EXP; ++e) acc[e] = 0.f;
  for (int d = lane; d < D_MODEL; d += 32) {
    float xv = x[(size_t)t * D_MODEL + d];
    #pragma unroll
    for (int e = 0; e < N_EXP; ++e)
      acc[e] = fmaf(xv, gw[e * D_MODEL + d], acc[e]);
  }
  #pragma unroll
  for (int e = 0; e < N_EXP; ++e) {
    float v = acc[e];
    #pragma unroll
    for (int off = 16; off > 0; off >>= 1) v += __shfl_xor(v, off, 32);
    acc[e] = v;
  }
  if (lane == 0) {
    float aff[N_EXP], score[N_EXP];
    #pragma unroll
    for (int e = 0; e < N_EXP; ++e) {
      aff[e]   = 1.f / (1.f + __expf(-acc[e]));
      score[e] = aff[e] + bias[e];
    }
    int pick[TOPK]; float pa[TOPK]; float s = 0.f;
    #pragma unroll
    for (int k = 0; k < TOPK; ++k) {
      int bi = 0; float bv = -1e30f;
      #pragma unroll
      for (int e = 0; e < N_EXP; ++e) {
        bool used = false;
        #pragma unroll
        for (int j = 0; j < TOPK; ++j) used |= (j < k) && (pick[j] == e);
        if (!used && score[e] > bv) { bv = score[e]; bi = e; }
      }
      pick[k] = bi; pa[k] = aff[bi]; s += aff[bi];
    }
    float inv = 1.f / (s + 1e-9f);
    #pragma unroll
    for (int k = 0; k < TOPK; ++k) {
      sel[t * TOPK + k]    = pick[k];
      gating[t * TOPK + k] = pa[k] * inv;
      atomicAdd(&counts[pick[k]], 1);
    }
  }
}

__global__ void prefix_kernel(const int* __restrict__ counts, int* __restrict__ basearr) {
  if (threadIdx.x == 0 && blockIdx.x == 0) {
    int s = 0;
    for (int e = 0; e < N_EXP; ++e) { basearr[e] = s; s += counts[e]; }
  }
}

__global__ __launch_bounds__(256) void fill_kernel(
    const int* __restrict__ sel, int* __restrict__ cnt2,
    const int* __restrict__ basearr, int* __restrict__ etok, int* __restrict__ slot_of)
{
  int i = blockIdx.x * blockDim.x + threadIdx.x;
  if (i >= NSLOT) return;
  int t = i / TOPK;
  int e = sel[i];
  int pos = atomicAdd(&cnt2[e], 1);
  int slot = basearr[e] + pos;
  etok[slot] = t;
  slot_of[i] = slot;
}

// ---------------- shared expert: writes d_out directly ----------------
__global__ __launch_bounds__(BLOCK) void shared_kernel(
    const float* __restrict__ x, const float* __restrict__ wgu,
    const float* __restrict__ wd, float* __restrict__ out)
{
  __shared__ TileSmem sm;
  if (threadIdx.x < MTILE) {
    int t = blockIdx.x * MTILE + threadIdx.x;
    sm.sTok[threadIdx.x] = t;
    sm.sDst[threadIdx.x] = t;
  }
  __syncthreads();
  tile_ffn(sm, x, wgu, wd, out);
}

// ---------------- routed experts: dense 32-row tiles per expert ----------------
__global__ __launch_bounds__(BLOCK) void routed_kernel(
    const float* __restrict__ x, const float* __restrict__ egu,
    const float* __restrict__ edn, float* __restrict__ slotbuf,
    const int* __restrict__ etok, const int* __restrict__ counts,
    const int* __restrict__ basearr)
{
  const int e    = blockIdx.y;
  const int tile = blockIdx.x;
  const int cnt  = counts[e];
  if (tile * MTILE >= cnt) return;        // uniform early exit (scalar branch)

  __shared__ TileSmem sm;
  if (threadIdx.x < MTILE) {
    int idx  = tile * MTILE + threadIdx.x;
    bool v   = idx < cnt;
    int slot = basearr[e] + idx;
    sm.sTok[threadIdx.x] = v ? etok[slot] : 0;
    sm.sDst[threadIdx.x] = v ? slot : -1; // padded rows: compute but never store
  }
  __syncthreads();
  tile_ffn(sm, x,
           egu + (size_t)e * 2 * D_FF * D_MODEL,
           edn + (size_t)e * D_MODEL * D_FF,
           slotbuf);
}

__global__ __launch_bounds__(256) void combine_kernel(
    float* __restrict__ out, const float* __restrict__ slotbuf,
    const int* __restrict__ slot_of, const float* __restrict__ gating)
{
  int i = blockIdx.x * blockDim.x + threadIdx.x;
  if (i >= T_TOK * D_MODEL) return;
  int t = i >> 9;
  int d = i & (D_MODEL - 1);
  float v = out[i];
  #pragma unroll
  for (int k = 0; k < TOPK; ++k) {
    int s = slot_of[t * TOPK + k];
    v = fmaf(gating[t * TOPK + k], slotbuf[(size_t)s * D_MODEL + d], v);
  }
  out[i] = v;
}

extern "C" void kernel_launch(void* const* d_in, const int* in_sizes, int n_in,
                              void* d_out, int out_size, void* d_ws, size_t ws_size,
                              hipStream_t stream) {
  const float* x    = (const float*)d_in[0];
  const float* gw   = (const float*)d_in[1];
  const float* bias = (const float*)d_in[2];
  const float* sgu  = (const float*)d_in[3];
  const float* sdn  = (const float*)d_in[4];
  const float* egu  = (const float*)d_in[5];
  const float* edn  = (const float*)d_in[6];
  float* out = (float*)d_out;

  char* w = (char*)d_ws;
  int*   counts  = (int*)(w);            // 16
  int*   cnt2    = (int*)(w + 64);       // 16
  int*   basearr = (int*)(w + 128);      // 16
  int*   sel     = (int*)(w + 256);      // NSLOT
  int*   slot_of = sel + NSLOT;          // NSLOT
  int*   etok    = slot_of + NSLOT;      // NSLOT + MTILE (read-padding)
  float* gating  = (float*)(etok + NSLOT + MTILE);
  size_t sb_off  = 256 + (size_t)(3 * NSLOT + MTILE + NSLOT) * sizeof(int);
  sb_off = (sb_off + 255) & ~(size_t)255;
  float* slotbuf = (float*)(w + sb_off); // NSLOT * D_MODEL floats (~12.6 MB)

  hipMemsetAsync(d_ws, 0, 256, stream);
  router_kernel<<<T_TOK / 8, 256, 0, stream>>>(x, gw, bias, sel, gating, counts);
  prefix_kernel<<<1, 32, 0, stream>>>(counts, basearr);
  fill_kernel<<<(NSLOT + 255) / 256, 256, 0, stream>>>(sel, cnt2, basearr, etok, slot_of);
  shared_kernel<<<T_TOK / MTILE, BLOCK, 0, stream>>>(x, sgu, sdn, out);
  routed_kernel<<<dim3(T_TOK / MTILE, N_EXP), BLOCK, 0, stream>>>(x, egu, edn, slotbuf,
                                                                  etok, counts, basearr);
  combine_kernel<<<(T_TOK * D_MODEL + 255) / 256, 256, 0, stream>>>(out, slotbuf,
                                                                    slot_of, gating);
}